// CIGLoss_79774722556578
// MI455X (gfx1250) — compile-verified
//
#include <hip/hip_runtime.h>

// Problem constants from the reference
#define NSEG   4000          // NUM_PATHS
#define PPI    500           // PATHS_PER_IMG
#define IMG_W  1024
#define IMG_HW (1024 * 1024)

#define BLOCK  256
#define NWAVES (BLOCK / 32)
#define TILES  16
#define SLAB   (BLOCK * TILES)   // 4096 pixels per block
#define SPAN   288               // LDS segment-accumulator window (~5 typical)

// ---- CDNA5 async global->LDS staging -------------------------------------
// Each lane stages its own 4B element to its own LDS slot and later reads
// only that slot, so completion only needs s_wait_asynccnt (no barrier).

__device__ __forceinline__ unsigned lds_of(const void* p) {
  // flat shared-memory pointer low 32 bits == wave-relative LDS byte address
  return (unsigned)(unsigned long long)p;
}

__device__ __forceinline__ void async_ld_b32(unsigned lds_addr, const void* gaddr) {
  asm volatile("global_load_async_to_lds_b32 %0, %1, off"
               :: "v"(lds_addr), "v"(gaddr) : "memory");
}

__device__ __forceinline__ void wait_async0() {
  asm volatile("s_wait_asynccnt 0x0" ::: "memory");
}

// ---- kernel 1: zero accumulators + output --------------------------------
__global__ void zero_acc(float* __restrict__ gsum, float* __restrict__ gcnt,
                         float* __restrict__ out) {
  int i = blockIdx.x * blockDim.x + threadIdx.x;
  if (i < NSEG) { gsum[i] = 0.0f; gcnt[i] = 0.0f; }
  if (i == 0) out[0] = 0.0f;
}

// ---- kernel 2: gather + per-segment sum/count ----------------------------
__global__ __launch_bounds__(BLOCK) void pass_sum(
    const float* __restrict__ img, const int* __restrict__ rows,
    const int* __restrict__ cols, const int* __restrict__ segs,
    float* __restrict__ gsum, float* __restrict__ gcnt,
    float* __restrict__ vcache, int n, int docache) {
  __shared__ int sR[2][BLOCK], sC[2][BLOCK], sS[2][BLOCK];
  __shared__ float lsum[SPAN], lcnt[SPAN];

  const int tid   = threadIdx.x;
  const int lane  = tid & 31;
  const int start = blockIdx.x * SLAB;
  if (start >= n) return;
  const int end   = min(start + SLAB, n);
  const int segLo = segs[start];
  const int segHi = segs[end - 1];
  const int span  = segHi - segLo + 1;       // sorted ids -> contiguous range
  const bool fits = (span <= SPAN);

  for (int s = tid; s < SPAN; s += BLOCK) { lsum[s] = 0.0f; lcnt[s] = 0.0f; }
  __syncthreads();

  const int ntiles = (end - start + BLOCK - 1) / BLOCK;

  // stage tile 0 (indices clamped -> always valid addresses, full EXEC)
  {
    int idx = min(start + tid, n - 1);
    async_ld_b32(lds_of(&sR[0][tid]), rows + idx);
    async_ld_b32(lds_of(&sC[0][tid]), cols + idx);
    async_ld_b32(lds_of(&sS[0][tid]), segs + idx);
  }

  for (int t = 0; t < ntiles; ++t) {
    wait_async0();                       // tile t staged
    const int buf = t & 1;
    const int r   = sR[buf][tid];
    const int c   = sC[buf][tid];
    const int seg = sS[buf][tid];

    if (t + 1 < ntiles) {                // stage tile t+1 into other buffer
      const int nb = (t + 1) & 1;
      int idx = min(start + (t + 1) * BLOCK + tid, n - 1);
      async_ld_b32(lds_of(&sR[nb][tid]), rows + idx);
      async_ld_b32(lds_of(&sC[nb][tid]), cols + idx);
      async_ld_b32(lds_of(&sS[nb][tid]), segs + idx);
    }

    const int idx    = start + t * BLOCK + tid;
    const bool valid = (idx < n);
    const int  b     = seg / PPI;                          // pix_batch
    const float vg   = img[b * IMG_HW + r * IMG_W + c];    // L2-resident gather
    if (valid && docache) vcache[idx] = vg;

    // ---- wave32 segmented reduction over equal-seg runs (sorted ids) ----
    // Run heads via one shuffle + ballot; value scan via 5 shuffles with the
    // run condition folded to a lane-distance test; counts via popcount.
    const int  prevseg  = __shfl_up(seg, 1, 32);
    const bool head     = (lane == 0) || (prevseg != seg);
    const unsigned headmask  = (unsigned)__ballot(head);
    const unsigned validmask = (unsigned)__ballot(valid);
    const unsigned lm_le     = (2u << lane) - 1u;          // lanes <= me
    const int headLane = 31 - __clz(headmask & lm_le);     // my run's first lane

    float runs = valid ? vg : 0.0f;
#pragma unroll
    for (int off = 1; off < 32; off <<= 1) {
      const float ps = __shfl_up(runs, off, 32);
      if (lane - off >= headLane) runs += ps;
    }

    const bool leader = (lane == 31) || (((headmask >> (lane + 1)) & 1u) != 0u);
    if (leader) {
      const unsigned runmask = lm_le & ~((1u << headLane) - 1u);
      const float runc = (float)__popc(validmask & runmask);
      if (runc > 0.0f) {                 // one atomic pair per run, not per pixel
        if (fits) {
          atomicAdd(&lsum[seg - segLo], runs);
          atomicAdd(&lcnt[seg - segLo], runc);
        } else {
          atomicAdd(&gsum[seg], runs);
          atomicAdd(&gcnt[seg], runc);
        }
      }
    }
  }
  __syncthreads();

  if (fits) {
    for (int s = tid; s < span; s += BLOCK) {
      const float cv = lcnt[s];
      if (cv != 0.0f) {
        atomicAdd(&gsum[segLo + s], lsum[s]);
        atomicAdd(&gcnt[segLo + s], cv);
      }
    }
  }
}

// ---- kernel 3: |val - mean| / count, reduce to scalar --------------------
__global__ __launch_bounds__(BLOCK) void pass_loss(
    const float* __restrict__ img, const int* __restrict__ rows,
    const int* __restrict__ cols, const int* __restrict__ segs,
    const float* __restrict__ gsum, const float* __restrict__ gcnt,
    const float* __restrict__ vcache, float* __restrict__ out,
    int n, int docache) {
  __shared__ int sR[2][BLOCK], sC[2][BLOCK], sS[2][BLOCK];
  __shared__ float lmean[SPAN], linv[SPAN];
  __shared__ float wred[NWAVES];

  const int tid   = threadIdx.x;
  const int lane  = tid & 31;
  const int start = blockIdx.x * SLAB;
  if (start >= n) return;
  const int end   = min(start + SLAB, n);
  const int segLo = segs[start];
  const int segHi = segs[end - 1];
  const int span  = segHi - segLo + 1;
  const bool fits = (span <= SPAN);

  if (fits) {
    for (int s = tid; s < span; s += BLOCK) {
      const float cc = fmaxf(gcnt[segLo + s], 1.0f);
      const float iv = 1.0f / cc;
      linv[s]  = iv;
      lmean[s] = gsum[segLo + s] * iv;
    }
  }
  __syncthreads();

  const int ntiles = (end - start + BLOCK - 1) / BLOCK;

  {
    int idx = min(start + tid, n - 1);
    if (!docache) {
      async_ld_b32(lds_of(&sR[0][tid]), rows + idx);
      async_ld_b32(lds_of(&sC[0][tid]), cols + idx);
    }
    async_ld_b32(lds_of(&sS[0][tid]), segs + idx);
  }

  float acc = 0.0f;
  for (int t = 0; t < ntiles; ++t) {
    wait_async0();
    const int buf = t & 1;
    int r = 0, c = 0;
    if (!docache) { r = sR[buf][tid]; c = sC[buf][tid]; }
    const int seg = sS[buf][tid];

    if (t + 1 < ntiles) {
      const int nb = (t + 1) & 1;
      int idx = min(start + (t + 1) * BLOCK + tid, n - 1);
      if (!docache) {
        async_ld_b32(lds_of(&sR[nb][tid]), rows + idx);
        async_ld_b32(lds_of(&sC[nb][tid]), cols + idx);
      }
      async_ld_b32(lds_of(&sS[nb][tid]), segs + idx);
    }

    const int idx    = start + t * BLOCK + tid;
    const bool valid = (idx < n);
    float v;
    if (docache) {
      v = vcache[min(idx, n - 1)];
    } else {
      const int b = seg / PPI;
      v = img[b * IMG_HW + r * IMG_W + c];
    }
    float m, iv;
    if (fits) {
      const int ls = seg - segLo;
      m  = lmean[ls];
      iv = linv[ls];
    } else {
      const float cc = fmaxf(gcnt[seg], 1.0f);
      iv = 1.0f / cc;
      m  = gsum[seg] * iv;
    }
    acc += valid ? fabsf(v - m) * iv : 0.0f;   // dev / counts[seg]
  }

  // wave32 butterfly reduce, then 8 partials summed by thread 0
#pragma unroll
  for (int off = 16; off > 0; off >>= 1) acc += __shfl_down(acc, off, 32);
  if (lane == 0) wred[tid >> 5] = acc;
  __syncthreads();
  if (tid == 0) {
    float s = 0.0f;
#pragma unroll
    for (int w = 0; w < NWAVES; ++w) s += wred[w];
    atomicAdd(out, s * 0.125f);                // / B
  }
}

// ---- host launcher -------------------------------------------------------
extern "C" void kernel_launch(void* const* d_in, const int* in_sizes, int n_in,
                              void* d_out, int out_size, void* d_ws, size_t ws_size,
                              hipStream_t stream) {
  const float* img  = (const float*)d_in[0];
  const int*   rows = (const int*)d_in[1];
  const int*   cols = (const int*)d_in[2];
  const int*   segs = (const int*)d_in[3];
  float*       out  = (float*)d_out;

  const int n = in_sizes[1];               // NUM_PIX

  float* gsum   = (float*)d_ws;            // [NSEG]
  float* gcnt   = gsum + NSEG;             // [NSEG]
  float* vcache = gsum + 2 * NSEG;         // [n] optional gather cache
  const int docache =
      (ws_size >= (size_t)(2 * NSEG + n) * sizeof(float)) ? 1 : 0;

  const int nblk = (n + SLAB - 1) / SLAB;

  zero_acc<<<(NSEG + BLOCK - 1) / BLOCK, BLOCK, 0, stream>>>(gsum, gcnt, out);
  pass_sum<<<nblk, BLOCK, 0, stream>>>(img, rows, cols, segs,
                                       gsum, gcnt, vcache, n, docache);
  pass_loss<<<nblk, BLOCK, 0, stream>>>(img, rows, cols, segs,
                                        gsum, gcnt, vcache, out, n, docache);
}